// AvgDistanceConv_80049600463401
// MI455X (gfx1250) — compile-verified
//
#include <hip/hip_runtime.h>

// Problem constants (from the reference).
constexpr int N_NODES = 100000;
constexpr int N_EDGES = 6400000;
constexpr int D_FEAT  = 128;

// Native clang vector types (required by __builtin_nontemporal_load/store).
typedef int   vint4   __attribute__((ext_vector_type(4)));
typedef float vfloat2 __attribute__((ext_vector_type(2)));

// ---------------------------------------------------------------------------
// Kernel 1: compact pos = h[:,0] into a dense, L2-resident 400KB array and
// zero the interleaved {sum, cnt} accumulator. Re-run on every launch so the
// kernel is graph-replay safe (harness does NOT re-poison between replays).
// ---------------------------------------------------------------------------
__global__ void __launch_bounds__(256)
avgdist_prep(const float* __restrict__ h,
             float* __restrict__ pos,        // [N_NODES]
             vfloat2* __restrict__ acc)      // [N_NODES] interleaved {sum, cnt}
{
    int i = blockIdx.x * 256 + threadIdx.x;
    if (i < N_NODES) {
        pos[i] = h[(size_t)i * D_FEAT];      // strided gather of column 0
        acc[i] = (vfloat2){0.0f, 0.0f};
    }
}

// ---------------------------------------------------------------------------
// Kernel 2: edge aggregation. 4 edges per thread.
//  - Index streams are single-use (51.2MB total): NT-hinted b128 loads so
//    they don't compete with the hot gather/accumulator set for cache.
//  - pos gathers hit the dense 400KB array (L2-resident, regular temporal).
//  - Two hardware global_atomic_add_f32 per edge, adjacent addresses ->
//    same L2 sector.
// ---------------------------------------------------------------------------
__global__ void __launch_bounds__(256)
avgdist_edges(const vint4* __restrict__ src4,
              const vint4* __restrict__ dst4,
              const float* __restrict__ pos,
              float* __restrict__ acc)       // acc[2*i]=sum, acc[2*i+1]=cnt
{
    int t = blockIdx.x * 256 + threadIdx.x;  // t in [0, N_EDGES/4)

    vint4 s = __builtin_nontemporal_load(&src4[t]);
    vint4 d = __builtin_nontemporal_load(&dst4[t]);

    float d0 = fabsf(pos[s.x] - pos[d.x]);
    float d1 = fabsf(pos[s.y] - pos[d.y]);
    float d2 = fabsf(pos[s.z] - pos[d.z]);
    float d3 = fabsf(pos[s.w] - pos[d.w]);

    // Hardware global_atomic_add_f32 (no CAS loop). Reference's segment_sum
    // is itself an unordered fp reduction, so fp-atomic ordering is fine.
    unsafeAtomicAdd(&acc[2 * (size_t)d.x],     d0);
    unsafeAtomicAdd(&acc[2 * (size_t)d.x + 1], 1.0f);
    unsafeAtomicAdd(&acc[2 * (size_t)d.y],     d1);
    unsafeAtomicAdd(&acc[2 * (size_t)d.y + 1], 1.0f);
    unsafeAtomicAdd(&acc[2 * (size_t)d.z],     d2);
    unsafeAtomicAdd(&acc[2 * (size_t)d.z + 1], 1.0f);
    unsafeAtomicAdd(&acc[2 * (size_t)d.w],     d3);
    unsafeAtomicAdd(&acc[2 * (size_t)d.w + 1], 1.0f);
}

// ---------------------------------------------------------------------------
// Kernel 3: finalize. out[i] = {pos[i], sum[i]/max(cnt[i],1)} as one b64
// store. acc is dead after this read -> NT (last-use) load; output is
// write-once -> NT store.
// ---------------------------------------------------------------------------
__global__ void __launch_bounds__(256)
avgdist_finalize(const float* __restrict__ pos,
                 const vfloat2* __restrict__ acc,
                 vfloat2* __restrict__ out)  // [N_NODES] rows of 2
{
    int i = blockIdx.x * 256 + threadIdx.x;
    if (i < N_NODES) {
        vfloat2 a = __builtin_nontemporal_load(&acc[i]);
        vfloat2 r = {pos[i], a.x / fmaxf(a.y, 1.0f)};
        __builtin_nontemporal_store(r, &out[i]);
    }
}

extern "C" void kernel_launch(void* const* d_in, const int* in_sizes, int n_in,
                              void* d_out, int out_size, void* d_ws, size_t ws_size,
                              hipStream_t stream)
{
    const float* h   = (const float*)d_in[0];
    const int*   src = (const int*)d_in[1];   // harness: integer -> const int*
    const int*   dst = (const int*)d_in[2];

    // Workspace layout (1.2 MB total):
    //   [0,           400KB)  dense pos[N_NODES]
    //   [400KB,       1.2MB)  interleaved acc[N_NODES] = {sum, cnt}
    float*   pos = (float*)d_ws;
    vfloat2* acc = (vfloat2*)((char*)d_ws + (size_t)N_NODES * sizeof(float));

    const int nodeBlocks = (N_NODES + 255) / 256;
    const int edgeBlocks = N_EDGES / (4 * 256);   // 6.4M / 1024 = 6250 exactly

    avgdist_prep<<<nodeBlocks, 256, 0, stream>>>(h, pos, acc);
    avgdist_edges<<<edgeBlocks, 256, 0, stream>>>((const vint4*)src, (const vint4*)dst,
                                                  pos, (float*)acc);
    avgdist_finalize<<<nodeBlocks, 256, 0, stream>>>(pos, acc, (vfloat2*)d_out);
}